// cooperative_pretraining_baseline_loss_12043088298402
// MI455X (gfx1250) — compile-verified
//
#include <hip/hip_runtime.h>

// Problem constants (match reference)
constexpr int B_ = 4, D_ = 64, H_ = 64, W_ = 64, C_ = 64;
constexpr int N_ = 20000, S_ = 2048;
#define INV_T 1.4285714285714286f /* 1/0.7 */

typedef float v2f __attribute__((ext_vector_type(2)));
typedef float v8f __attribute__((ext_vector_type(8)));

__global__ void zero_out_kernel(float* out) { out[0] = 0.0f; }

// inv_perm[b, shuf[b,i]] = i   (shuf is a permutation of 0..N-1 per batch)
__global__ void inv_perm_kernel(const int* __restrict__ shuf,
                                int* __restrict__ invPerm) {
  int i = blockIdx.x * blockDim.x + threadIdx.x;
  if (i >= B_ * N_) return;
  int b = i / N_, j = i - b * N_;
  invPerm[b * N_ + shuf[i]] = j;
}

// One wave per sampled point: gather 64-f32 feature row from the dense grid,
// L2-normalize (wave-wide shfl reduction), store contiguously for WMMA use.
__global__ void __launch_bounds__(256)
gather_norm_kernel(const float* __restrict__ dv, const int* __restrict__ vcoor,
                   const float* __restrict__ df, const int* __restrict__ fcoor,
                   const int* __restrict__ invPerm, const int* __restrict__ samp,
                   float* __restrict__ Vf, float* __restrict__ Ff) {
  int lane = threadIdx.x & 31;
  int wave = threadIdx.x >> 5;
  int gs = blockIdx.x * 8 + wave;          // global sample id in [0, B*S)
  int b = gs / S_, s = gs - b * S_;
  int sp = samp[b * S_ + s];

  // vehicle feature: coors[b, sp, :] // 8
  const int* vc = vcoor + ((long)b * N_ + sp) * 3;
  int vx = vc[0] >> 3, vy = vc[1] >> 3, vz = vc[2] >> 3;
  const float* vs = dv + ((((long)b * D_ + vx) * H_ + vy) * W_ + vz) * C_;
  float2 v = *(const float2*)(vs + lane * 2);
  float ss = v.x * v.x + v.y * v.y;
  for (int m = 1; m < 32; m <<= 1) ss += __shfl_xor(ss, m, 32);
  float inv = 1.0f / fmaxf(sqrtf(ss), 1e-12f);
  float2 vn; vn.x = v.x * inv; vn.y = v.y * inv;
  *(float2*)(Vf + ((long)b * S_ + s) * C_ + lane * 2) = vn;

  // fusion feature: coors[b, inv_perm[sp], :] // 8
  int fid = invPerm[b * N_ + sp];
  const int* fc = fcoor + ((long)b * N_ + fid) * 3;
  int fx = fc[0] >> 3, fy = fc[1] >> 3, fz = fc[2] >> 3;
  const float* fs = df + ((((long)b * D_ + fx) * H_ + fy) * W_ + fz) * C_;
  float2 f = *(const float2*)(fs + lane * 2);
  float fss = f.x * f.x + f.y * f.y;
  for (int m = 1; m < 32; m <<= 1) fss += __shfl_xor(fss, m, 32);
  float finv = 1.0f / fmaxf(sqrtf(fss), 1e-12f);
  float2 fn; fn.x = f.x * finv; fn.y = f.y * finv;
  *(float2*)(Ff + ((long)b * S_ + s) * C_ + lane * 2) = fn;
}

// grid = B * (S/16) blocks, 256 threads (8 waves). Block owns 16 rows of V;
// each wave strides over 16-column tiles of F with register double-buffered
// B-fragments: prefetch tile t+8 while the 16-deep WMMA chain eats tile t.
__global__ void __launch_bounds__(256)
contrastive_loss_kernel(const float* __restrict__ Vf,
                        const float* __restrict__ Ff,
                        float* __restrict__ out) {
  constexpr int TILES = S_ / 16;  // 128
  constexpr int NW = 8;           // waves per block
  __shared__ float rowsum_s[NW][16];
  __shared__ float diag_s[16];

  int m = blockIdx.x % TILES;     // row-tile index
  int b = blockIdx.x / TILES;
  int lane = threadIdx.x & 31;
  int wave = threadIdx.x >> 5;
  int row0 = m * 16;

  int half = lane >> 4;           // 0: lanes 0-15 (K 0..1), 1: lanes 16-31 (K 2..3)
  int l15 = lane & 15;
  int koff = half * 2;

  const float* Vb = Vf + (long)b * S_ * C_;
  const float* Fb = Ff + (long)b * S_ * C_;

  // Preload A fragments for this block's 16 rows: a[k] covers K = 4k..4k+3
  // (ISA 32-bit A layout: VGPR0=K{0,2}, VGPR1=K{1,3}).
  v2f a[16];
#pragma unroll
  for (int k = 0; k < 16; ++k)
    a[k] = *(const v2f*)(Vb + (long)(row0 + l15) * C_ + k * 4 + koff);

  float rowAcc[8];
#pragma unroll
  for (int r = 0; r < 8; ++r) rowAcc[r] = 0.0f;
  float diagAcc = 0.0f;

  auto loadB = [&](int t, v2f* dst) {
    const float* base = Fb + (long)(t * 16 + l15) * C_ + koff;
#pragma unroll
    for (int k = 0; k < 16; ++k) dst[k] = *(const v2f*)(base + k * 4);
  };

  auto computeTile = [&](int t, const v2f* bf) {
    int col0 = t * 16;
    v8f c = {0.f, 0.f, 0.f, 0.f, 0.f, 0.f, 0.f, 0.f};
#pragma unroll
    for (int k = 0; k < 16; ++k)
      c = __builtin_amdgcn_wmma_f32_16x16x4_f32(false, a[k], false, bf[k],
                                                (short)0, c, false, false);
#pragma unroll
    for (int r = 0; r < 8; ++r) {
      float val = c[r];               // s[row0+M][col0+N], M=r+8*half, N=l15
      rowAcc[r] += __expf(val * INV_T);
      if (col0 == row0) {
        bool isDiag = (l15 == r + half * 8);   // N == M
        diagAcc += isDiag ? val : 0.0f;
      }
    }
  };

  // Software pipeline over this wave's 16 tiles: t = wave + i*NW, i = 0..15.
  v2f b0[16], b1[16];
  loadB(wave, b0);
#pragma unroll 1
  for (int i = 0; i < TILES / NW; i += 2) {
    int t0 = wave + i * NW;
    loadB(t0 + NW, b1);               // prefetch odd tile
    computeTile(t0, b0);
    if (i + 2 < TILES / NW) loadB(t0 + 2 * NW, b0);  // prefetch next even tile
    computeTile(t0 + NW, b1);
  }

  // Reduce rowAcc over each 16-lane half (rows r on lanes 0-15, r+8 on 16-31)
#pragma unroll
  for (int r = 0; r < 8; ++r) {
    float v = rowAcc[r];
    v += __shfl_xor(v, 1, 32);
    v += __shfl_xor(v, 2, 32);
    v += __shfl_xor(v, 4, 32);
    v += __shfl_xor(v, 8, 32);
    rowAcc[r] = v;
  }
  if (lane == 0) {
#pragma unroll
    for (int r = 0; r < 8; ++r) rowsum_s[wave][r] = rowAcc[r];
  }
  if (lane == 16) {
#pragma unroll
    for (int r = 0; r < 8; ++r) rowsum_s[wave][8 + r] = rowAcc[r];
  }
  // Diagonal tile (col0 == row0) belongs to wave (m & 7); diag of row rr sits
  // on lane rr (rr<8) or lane 16+rr (rr>=8).
  if (wave == (m & (NW - 1))) {
    if (lane < 8) diag_s[lane] = diagAcc;
    else if (lane >= 24) diag_s[lane - 16] = diagAcc;
  }
  __syncthreads();

  if (wave == 0 && lane < 16) {
    float total = 0.0f;
#pragma unroll
    for (int w = 0; w < NW; ++w) total += rowsum_s[w][lane];
    float loss = __logf(total) - diag_s[lane] * INV_T;
    loss += __shfl_xor(loss, 1, 32);
    loss += __shfl_xor(loss, 2, 32);
    loss += __shfl_xor(loss, 4, 32);
    loss += __shfl_xor(loss, 8, 32);
    if (lane == 0) atomicAdd(out, loss * (1.0f / (float)S_));
  }
}

extern "C" void kernel_launch(void* const* d_in, const int* in_sizes, int n_in,
                              void* d_out, int out_size, void* d_ws, size_t ws_size,
                              hipStream_t stream) {
  const float* dv    = (const float*)d_in[0];  // dense_vehicle_voxel_feats (B,D,H,W,C)
  const int*   vcoor = (const int*)d_in[1];    // vehicle_voxels_coors (B,N,3)
  const float* df    = (const float*)d_in[2];  // dense_fusion_voxel_feats
  const int*   fcoor = (const int*)d_in[3];    // fusion_voxels_coors
  const int*   shuf  = (const int*)d_in[4];    // fusion_points_shuffled_idx (B,N)
  // d_in[5] = vehicle_points (unused by the loss)
  const int*   samp  = (const int*)d_in[6];    // sampled_idx (B,S)
  float* out = (float*)d_out;

  // Workspace layout (256B aligned): invPerm | Vf | Ff
  char* ws = (char*)d_ws;
  int* invPerm = (int*)ws;
  size_t off = ((size_t)B_ * N_ * sizeof(int) + 255) & ~(size_t)255;
  float* Vf = (float*)(ws + off);
  float* Ff = Vf + (size_t)B_ * S_ * C_;

  zero_out_kernel<<<1, 1, 0, stream>>>(out);
  inv_perm_kernel<<<(B_ * N_ + 255) / 256, 256, 0, stream>>>(shuf, invPerm);
  gather_norm_kernel<<<(B_ * S_) / 8, 256, 0, stream>>>(dv, vcoor, df, fcoor,
                                                        invPerm, samp, Vf, Ff);
  contrastive_loss_kernel<<<B_ * (S_ / 16), 256, 0, stream>>>(Vf, Ff, out);
}